// Model_8263517077665
// MI455X (gfx1250) — compile-verified
//
#include <hip/hip_runtime.h>
#include <stdint.h>
#include <stddef.h>

// ---------------------------------------------------------------------------
// CDNA5 (gfx1250, wave32) sparse submanifold 3D UNet.
// Core op: 27-offset rulebook conv as async global->LDS gather (double
// buffered, ASYNCcnt-pipelined) feeding v_wmma_f32_16x16x4_f32 tiles.
// ---------------------------------------------------------------------------

typedef __attribute__((ext_vector_type(2))) float v2f;
typedef __attribute__((ext_vector_type(8))) float v8f;

#define BN_EPS 1e-4f
#define MAX_CB 7      // cout <= 112 -> up to 7 column blocks of 16
#define MAXC   224    // max channel count anywhere (2*96=192, +margin)

// --------------------------- utility kernels --------------------------------

__global__ void zero_f(float* __restrict__ p, size_t n) {
  size_t i = (size_t)blockIdx.x * blockDim.x + threadIdx.x;
  size_t st = (size_t)gridDim.x * blockDim.x;
  for (; i < n; i += st) p[i] = 0.0f;
}

// Column sums / sum-of-squares over [N, C] row-major, via LDS float atomics
// then one global float-atomic per channel per block.
__global__ void bn_stats(const float* __restrict__ x, int N, int C,
                         float* __restrict__ sum, float* __restrict__ sumsq) {
  __shared__ float ls[MAXC];
  __shared__ float lq[MAXC];
  for (int c = threadIdx.x; c < C; c += blockDim.x) { ls[c] = 0.f; lq[c] = 0.f; }
  __syncthreads();
  size_t tot = (size_t)N * (size_t)C;
  size_t i = (size_t)blockIdx.x * blockDim.x + threadIdx.x;
  size_t st = (size_t)gridDim.x * blockDim.x;
  for (; i < tot; i += st) {
    float v = x[i];
    int c = (int)(i % (size_t)C);
    atomicAdd(&ls[c], v);
    atomicAdd(&lq[c], v * v);
  }
  __syncthreads();
  for (int c = threadIdx.x; c < C; c += blockDim.x) {
    atomicAdd(&sum[c], ls[c]);
    atomicAdd(&sumsq[c], lq[c]);
  }
}

// scale = g * rsqrt(var + eps); shift = b - mu * scale  (biased var, as jnp.var)
__global__ void bn_finalize(const float* __restrict__ sum, const float* __restrict__ sumsq,
                            const float* __restrict__ g, const float* __restrict__ b,
                            int N, int C,
                            float* __restrict__ scale, float* __restrict__ shift) {
  int c = threadIdx.x + blockIdx.x * blockDim.x;
  if (c >= C) return;
  float inv = 1.0f / (float)N;
  float mu = sum[c] * inv;
  float var = sumsq[c] * inv - mu * mu;
  float s = g[c] * rsqrtf(var + BN_EPS);
  scale[c] = s;
  shift[c] = b[c] - mu * s;
}

__global__ void concat_k(const float* __restrict__ a, const float* __restrict__ b,
                         float* __restrict__ y, int N, int c1, int c2) {
  int ct = c1 + c2;
  size_t tot = (size_t)N * ct;
  size_t i = (size_t)blockIdx.x * blockDim.x + threadIdx.x;
  size_t st = (size_t)gridDim.x * blockDim.x;
  for (; i < tot; i += st) {
    size_t n = i / (size_t)ct;
    int j = (int)(i - n * (size_t)ct);
    y[i] = (j < c1) ? a[n * c1 + j] : b[n * c2 + (j - c1)];
  }
}

// ------------------- submanifold conv: async gather + WMMA ------------------
//
// One wave32 per 16-row output tile. For each of the 27 offsets:
//   AFF path (all BN convs, cin==CINP multiple of 16):
//     stage k+1's 16 neighbor rows into LDS with GLOBAL_LOAD_ASYNC_TO_LDS_B128
//     (double-buffered), s_wait_asynccnt, then K-march offset k with
//     V_WMMA_F32_16X16X4_F32, fusing BN scale/shift + ReLU at A-fragment read.
//   !AFF path (conv0, cin=3): synchronous scalar gather with zero padding.
//
// Invalid neighbors clamp to row 0 at issue (deterministic ASYNCcnt) and are
// zero-masked at fragment read from the rulebook entry.
//
// A 16x4 f32 fragment (ISA 7.12.2): lane l holds row m=l&15, channels
// base + (l>>4)*2 + {0,1} in its 2 VGPRs. B 4x16 mirrors with M<->N.
// C/D 16x16 f32: VGPR j holds row j (lanes 0-15) / row j+8 (lanes 16-31).
// LDS row stride CINP+4: 16B-aligned for b128 async writes and ==4 (mod 16)
// so the 16 row addresses of an A-column read map to 16 distinct banks.

template <int CINP, bool AFF>
__global__ __launch_bounds__(32)
void subm_conv_wmma(const float* __restrict__ x, const int* __restrict__ nbr,
                    const float* __restrict__ W,           // [27][cin][cout]
                    const float* __restrict__ scale,       // [cin] or null
                    const float* __restrict__ shift,       // [cin] or null
                    const float* __restrict__ addsrc,      // [N,cout] or null
                    float* __restrict__ y,                 // [N,cout]
                    int N, int cin, int cout) {
  constexpr int LDA   = CINP + 4;
  constexpr int TRIPS = (4 * CINP) / 32;    // b128 chunks per lane per stage
  __shared__ __align__(16) float ldsA[2 * 16 * LDA];
  __shared__ float ldsS[2 * CINP];          // interleaved (scale, shift)
  __shared__ int   ldsN[16 * 27];
  volatile float* vA = ldsA;                // async engine writes are invisible
                                            // to the compiler -> volatile reads
  const int lane = threadIdx.x;
  const int row0 = blockIdx.x * 16;

  for (int e = lane; e < 16 * 27; e += 32) {
    int rl = e / 27, k = e - rl * 27;
    int r = row0 + rl;
    ldsN[e] = (r < N) ? nbr[(size_t)r * 27 + k] : -1;
  }
  if (AFF) {
    for (int c = lane; c < CINP; c += 32) {
      ldsS[2 * c]     = scale[c];
      ldsS[2 * c + 1] = shift[c];
    }
  }
  __syncthreads();

  v8f acc[MAX_CB] = {};
  const int ncb   = cout >> 4;              // cout is a multiple of 16
  const int m     = lane & 15;
  const int khalf = (lane >> 4) * 2;

  if (AFF) {
    // ---- issue one stage of async gathers into buffer (k&1) ----
    auto issue = [&](int k) {
      const int base = (k & 1) * 16 * LDA;
#pragma unroll
      for (int t = 0; t < TRIPS; ++t) {
        int e  = t * 32 + lane;             // chunk id in [0, 4*CINP)
        int rl = e / (CINP / 4);
        int cq = e - rl * (CINP / 4);
        int src = ldsN[rl * 27 + k];
        int se = (src < 0) ? 0 : src;       // clamp: deterministic issue count
        const float* ga = x + ((size_t)se * CINP + (size_t)cq * 4);
        unsigned la = (unsigned)(uintptr_t)&ldsA[base + rl * LDA + cq * 4];
        asm volatile("global_load_async_to_lds_b128 %0, %1, off"
                     :: "v"(la), "v"(ga) : "memory");
      }
    };

    issue(0);
    for (int k = 0; k < 27; ++k) {
      if (k + 1 < 27) {
        issue(k + 1);
        // only stage k+1's TRIPS loads may remain outstanding -> stage k landed
        asm volatile("s_wait_asynccnt %0" :: "i"(TRIPS) : "memory");
      } else {
        asm volatile("s_wait_asynccnt 0" ::: "memory");
      }
      const int   base  = (k & 1) * 16 * LDA;
      const int   srcm  = ldsN[m * 27 + k];
      const float vmask = (srcm >= 0) ? 1.0f : 0.0f;
      const float* Wk = W + (size_t)k * CINP * cout;
      for (int kk = 0; kk < CINP; kk += 4) {
        int c = kk + khalf;
        float r0 = vA[base + m * LDA + c];
        float r1 = vA[base + m * LDA + c + 1];
        v2f A;
        A.x = fmaxf(r0 * ldsS[2 * c]     + ldsS[2 * c + 1], 0.0f) * vmask;
        A.y = fmaxf(r1 * ldsS[2 * c + 2] + ldsS[2 * c + 3], 0.0f) * vmask;
#pragma unroll
        for (int cb = 0; cb < MAX_CB; ++cb) {
          if (cb < ncb) {
            v2f B;
            B.x = Wk[(size_t)c       * cout + cb * 16 + m];
            B.y = Wk[(size_t)(c + 1) * cout + cb * 16 + m];
            acc[cb] = __builtin_amdgcn_wmma_f32_16x16x4_f32(
                false, A, false, B, (short)0, acc[cb], false, false);
          }
        }
      }
    }
  } else {
    // ---- synchronous scalar gather (conv0: cin=3 padded to 4, no BN) ----
    for (int k = 0; k < 27; ++k) {
      __syncthreads();
      for (int e = lane; e < 16 * CINP; e += 32) {
        int rl = e / CINP, c = e - rl * CINP;
        int src = ldsN[rl * 27 + k];
        float v = 0.0f;
        if (src >= 0 && c < cin) v = x[(size_t)src * cin + c];
        ldsA[rl * LDA + c] = v;
      }
      __syncthreads();
      const float* Wk = W + (size_t)k * cin * cout;
      for (int kk = 0; kk < CINP; kk += 4) {
        int c = kk + khalf;
        v2f A;
        A.x = ldsA[m * LDA + c];
        A.y = ldsA[m * LDA + c + 1];
#pragma unroll
        for (int cb = 0; cb < MAX_CB; ++cb) {
          if (cb < ncb) {
            v2f B;
            B.x = (c     < cin) ? Wk[(size_t)c       * cout + cb * 16 + m] : 0.0f;
            B.y = (c + 1 < cin) ? Wk[(size_t)(c + 1) * cout + cb * 16 + m] : 0.0f;
            acc[cb] = __builtin_amdgcn_wmma_f32_16x16x4_f32(
                false, A, false, B, (short)0, acc[cb], false, false);
          }
        }
      }
    }
  }

  // epilogue: D layout -> rows, optional residual add
#pragma unroll
  for (int cb = 0; cb < MAX_CB; ++cb) {
    if (cb < ncb) {
#pragma unroll
      for (int j = 0; j < 8; ++j) {
        int mr = j + ((lane >> 4) << 3);
        int r = row0 + mr;
        if (r < N) {
          int col = cb * 16 + m;
          float v = acc[cb][j];
          if (addsrc) v += addsrc[(size_t)r * cout + col];
          y[(size_t)r * cout + col] = v;
        }
      }
    }
  }
}

// ------------------- stride-2 down / up convs -------------------------------

__global__ void down_conv_k(const float* __restrict__ x, const int* __restrict__ parent,
                            const int* __restrict__ off, const float* __restrict__ W, // [8][cin][cout]
                            const float* __restrict__ scale, const float* __restrict__ shift,
                            float* __restrict__ y, int N, int cin, int cout) {
  __shared__ float xr[MAXC];
  int n = blockIdx.x;
  for (int c = threadIdx.x; c < cin; c += blockDim.x) {
    float v = x[(size_t)n * cin + c];
    xr[c] = fmaxf(v * scale[c] + shift[c], 0.0f);
  }
  __syncthreads();
  int o = off[n], p = parent[n];
  const float* Wo = W + (size_t)o * cin * cout;
  for (int d = threadIdx.x; d < cout; d += blockDim.x) {
    float s = 0.0f;
    for (int c = 0; c < cin; ++c) s += xr[c] * Wo[(size_t)c * cout + d];
    atomicAdd(&y[(size_t)p * cout + d], s);   // segment_sum to parent
  }
}

__global__ void up_conv_k(const float* __restrict__ x, const int* __restrict__ parent,
                          const int* __restrict__ off, const float* __restrict__ W,  // [8][cin][cout]
                          const float* __restrict__ scale, const float* __restrict__ shift,
                          float* __restrict__ y, int N, int cin, int cout) {
  __shared__ float xr[MAXC];
  int n = blockIdx.x;
  int p = parent[n];
  for (int c = threadIdx.x; c < cin; c += blockDim.x) {
    float v = x[(size_t)p * cin + c];
    xr[c] = fmaxf(v * scale[c] + shift[c], 0.0f);
  }
  __syncthreads();
  int o = off[n];
  const float* Wo = W + (size_t)o * cin * cout;
  for (int d = threadIdx.x; d < cout; d += blockDim.x) {
    float s = 0.0f;
    for (int c = 0; c < cin; ++c) s += xr[c] * Wo[(size_t)c * cout + d];
    y[(size_t)n * cout + d] = s;
  }
}

// dense per-row matmul: y[n] = (aff?relu(x*scale+shift):x) @ W (+bias)
__global__ void rowmm_k(const float* __restrict__ x, const float* __restrict__ W,
                        const float* __restrict__ bias,
                        const float* __restrict__ scale, const float* __restrict__ shift,
                        float* __restrict__ y, int N, int cin, int cout, int use_aff) {
  __shared__ float xr[MAXC];
  int n = blockIdx.x;
  for (int c = threadIdx.x; c < cin; c += blockDim.x) {
    float v = x[(size_t)n * cin + c];
    if (use_aff) v = fmaxf(v * scale[c] + shift[c], 0.0f);
    xr[c] = v;
  }
  __syncthreads();
  for (int d = threadIdx.x; d < cout; d += blockDim.x) {
    float s = bias ? bias[d] : 0.0f;
    for (int c = 0; c < cin; ++c) s += xr[c] * W[(size_t)c * cout + d];
    y[(size_t)n * cout + d] = s;
  }
}

// --------------------------- host orchestration -----------------------------

struct BlockP { const float *g1, *b1, *w1, *g2, *b2, *w2, *nin; };

static inline int gridFor(size_t n, int block, int cap) {
  size_t b = (n + (size_t)block - 1) / (size_t)block;
  if (b > (size_t)cap) b = (size_t)cap;
  if (b < 1) b = 1;
  return (int)b;
}

static void launch_bn(hipStream_t s, const float* x, int N, int C,
                      const float* g, const float* b,
                      float* sum, float* scale, float* shift) {
  zero_f<<<1, 256, 0, s>>>(sum, 512);   // sum[256] | sumsq[256]
  bn_stats<<<gridFor((size_t)N * C, 256, 2048), 256, 0, s>>>(x, N, C, sum, sum + 256);
  bn_finalize<<<1, 256, 0, s>>>(sum, sum + 256, g, b, N, C, scale, shift);
}

static void launch_subm(hipStream_t s, float* y, const float* x, const int* nbr,
                        const float* W, int N, int cin, int cout,
                        const float* scale, const float* shift, const float* add) {
  int cinp = (cin + 3) & ~3;            // pad cin=3 (conv0) to 4
  dim3 g((N + 15) / 16);
#define SUBM_CASE(CP, AF) \
  case CP: subm_conv_wmma<CP, AF><<<g, 32, 0, s>>>(x, nbr, W, scale, shift, add, y, N, cin, cout); break;
  switch (cinp) {
    SUBM_CASE(4,   false)   // conv0
    SUBM_CASE(16,  true)
    SUBM_CASE(32,  true)
    SUBM_CASE(48,  true)
    SUBM_CASE(64,  true)
    SUBM_CASE(80,  true)
    SUBM_CASE(96,  true)
    SUBM_CASE(112, true)
    SUBM_CASE(128, true)
    SUBM_CASE(160, true)
    SUBM_CASE(192, true)
    default: break;
  }
#undef SUBM_CASE
}

// res_block: out = shortcut(x) + conv2(bnrelu(conv1(bnrelu(x))))
// shortcut = x (a==b) or x @ nin (a!=b). Safe for out == x_in (conv2 gathers
// from Tmid; the residual read/write is per-element in the epilogue).
static void run_block(hipStream_t s, const float* x_in, int a, int b_, float* out,
                      const int* nbr, int N, const BlockP& p,
                      float* Tmid, float* Tsc, float* sum, float* scale, float* shift) {
  const float* addp;
  if (p.nin) {
    rowmm_k<<<N, 128, 0, s>>>(x_in, p.nin, nullptr, nullptr, nullptr, Tsc, N, a, b_, 0);
    addp = Tsc;
  } else {
    addp = x_in;
  }
  launch_bn(s, x_in, N, a, p.g1, p.b1, sum, scale, shift);
  launch_subm(s, Tmid, x_in, nbr, p.w1, N, a, b_, scale, shift, nullptr);
  launch_bn(s, Tmid, N, b_, p.g2, p.b2, sum, scale, shift);
  launch_subm(s, out, Tmid, nbr, p.w2, N, b_, b_, scale, shift, addp);
}

extern "C" void kernel_launch(void* const* d_in, const int* in_sizes, int n_in,
                              void* d_out, int out_size, void* d_ws, size_t ws_size,
                              hipStream_t stream) {
  (void)n_in; (void)out_size; (void)ws_size;

  // ---- parse inputs in setup_inputs() dict-insertion order ----
  int i = 0;
  const float* feats = (const float*)d_in[i];
  const int N0 = in_sizes[i] / 3; i++;
  const float* conv0_w = (const float*)d_in[i++];

  BlockP enc[7][2]; BlockP dec[6][2];
  const float *bnd_g[6], *bnd_b[6], *down_w[6], *bnu_g[6], *bnu_b[6], *up_w[6];
  for (int l = 0; l < 7; ++l) {
    for (int blk = 0; blk < 2; ++blk) {
      BlockP& p = enc[l][blk];
      p.g1 = (const float*)d_in[i++]; p.b1 = (const float*)d_in[i++];
      p.w1 = (const float*)d_in[i++];
      p.g2 = (const float*)d_in[i++]; p.b2 = (const float*)d_in[i++];
      p.w2 = (const float*)d_in[i++]; p.nin = nullptr;
    }
    if (l < 6) {
      bnd_g[l] = (const float*)d_in[i++]; bnd_b[l] = (const float*)d_in[i++];
      down_w[l] = (const float*)d_in[i++];
      bnu_g[l] = (const float*)d_in[i++]; bnu_b[l] = (const float*)d_in[i++];
      up_w[l] = (const float*)d_in[i++];
      BlockP& q0 = dec[l][0];                               // block(2*pl, pl) + nin
      q0.g1 = (const float*)d_in[i++]; q0.b1 = (const float*)d_in[i++];
      q0.w1 = (const float*)d_in[i++];
      q0.g2 = (const float*)d_in[i++]; q0.b2 = (const float*)d_in[i++];
      q0.w2 = (const float*)d_in[i++];
      q0.nin = (const float*)d_in[i++];
      BlockP& q1 = dec[l][1];                               // block(pl, pl)
      q1.g1 = (const float*)d_in[i++]; q1.b1 = (const float*)d_in[i++];
      q1.w1 = (const float*)d_in[i++];
      q1.g2 = (const float*)d_in[i++]; q1.b2 = (const float*)d_in[i++];
      q1.w2 = (const float*)d_in[i++]; q1.nin = nullptr;
    }
  }
  const float* bnf_g = (const float*)d_in[i++];
  const float* bnf_b = (const float*)d_in[i++];
  const float* lin_w = (const float*)d_in[i++];
  const float* lin_b = (const float*)d_in[i++];

  const int* nbr[7]; int Nl[7];
  for (int l = 0; l < 7; ++l) { nbr[l] = (const int*)d_in[i]; Nl[l] = in_sizes[i] / 27; i++; }
  const int* parent[6]; for (int l = 0; l < 6; ++l) parent[l] = (const int*)d_in[i++];
  const int* offp[6];   for (int l = 0; l < 6; ++l) offp[l]   = (const int*)d_in[i++];

  // ---- workspace layout (bump allocator, floats) ----
  float* ws = (float*)d_ws;
  float* sum   = ws;            // 512: sum|sumsq
  float* scale = ws + 512;      // 256
  float* shift = ws + 768;      // 256
  size_t pos = 1024;
  int pl[7]; for (int l = 0; l < 7; ++l) pl[l] = 16 * (l + 1);
  float* X[7];
  size_t M = 0;
  for (int l = 0; l < 7; ++l) {
    X[l] = ws + pos;
    pos += (size_t)Nl[l] * pl[l];
    size_t m = (size_t)Nl[l] * 2 * pl[l];
    if (m > M) M = m;
  }
  float* T0 = ws + pos; pos += M;   // up-conv result / conv temp
  float* T1 = ws + pos; pos += M;   // concat buffer [N, 2*pl]
  float* T2 = ws + pos; pos += M;   // NiN shortcut

  // ---- conv0: feats (cin=3) -> X[0], no BN ----
  launch_subm(stream, X[0], feats, nbr[0], conv0_w, N0, 3, 16, nullptr, nullptr, nullptr);

  // ---- encoder ----
  for (int l = 0; l < 7; ++l) {
    run_block(stream, X[l], pl[l], pl[l], X[l], nbr[l], Nl[l], enc[l][0], T0, T2, sum, scale, shift);
    run_block(stream, X[l], pl[l], pl[l], X[l], nbr[l], Nl[l], enc[l][1], T0, T2, sum, scale, shift);
    if (l < 6) {
      launch_bn(stream, X[l], Nl[l], pl[l], bnd_g[l], bnd_b[l], sum, scale, shift);
      size_t nz = (size_t)Nl[l + 1] * pl[l + 1];
      zero_f<<<gridFor(nz, 256, 4096), 256, 0, stream>>>(X[l + 1], nz);
      down_conv_k<<<Nl[l], 128, 0, stream>>>(X[l], parent[l], offp[l], down_w[l],
                                             scale, shift, X[l + 1], Nl[l], pl[l], pl[l + 1]);
    }
  }

  // ---- decoder ----
  for (int l = 5; l >= 0; --l) {
    launch_bn(stream, X[l + 1], Nl[l + 1], pl[l + 1], bnu_g[l], bnu_b[l], sum, scale, shift);
    up_conv_k<<<Nl[l], 128, 0, stream>>>(X[l + 1], parent[l], offp[l], up_w[l],
                                         scale, shift, T0, Nl[l], pl[l + 1], pl[l]);
    size_t tot = (size_t)Nl[l] * 2 * pl[l];
    concat_k<<<gridFor(tot, 256, 4096), 256, 0, stream>>>(X[l], T0, T1, Nl[l], pl[l], pl[l]);
    run_block(stream, T1, 2 * pl[l], pl[l], X[l], nbr[l], Nl[l], dec[l][0], T0, T2, sum, scale, shift);
    run_block(stream, X[l], pl[l], pl[l], X[l], nbr[l], Nl[l], dec[l][1], T0, T2, sum, scale, shift);
  }

  // ---- final BN+ReLU + linear (16 -> 20) ----
  launch_bn(stream, X[0], N0, 16, bnf_g, bnf_b, sum, scale, shift);
  rowmm_k<<<N0, 64, 0, stream>>>(X[0], lin_w, lin_b, scale, shift,
                                 (float*)d_out, N0, 16, 20, 1);
}